// UltraGCN_31585189495423
// MI455X (gfx1250) — compile-verified
//
#include <hip/hip_runtime.h>
#include <hip/hip_bf16.h>
#include <math.h>

typedef float v2f __attribute__((ext_vector_type(2)));
typedef float v8f __attribute__((ext_vector_type(8)));

#define UG_BATCH     4096
#define UG_DIM       64
#define UG_NUM_NEG   300
#define UG_NUM_NBR   10
#define UG_NEG_TILES 19          // ceil(300/16); tiles 0..17 full, tile 18 has 12 rows
#define UG_W1        1e-6f
#define UG_GAMMA     1e-4f
#define UG_LAMBDA    5e-4f
#define UG_MAIN_BLOCKS 512       // 4096 rows / 8 waves per block
#define UG_NORM_BLOCKS 600

// log1p(exp(-|x|)) + relu(-x)  ==  softplus(-x)  ==  -log_sigmoid(x)
__device__ __forceinline__ float ug_log_term(float x) {
    return log1pf(expf(-fabsf(x))) + fmaxf(-x, 0.0f);
}

// C/D layout (ISA 7.12.2): VGPR r, lanes 0-15 -> M=r, lanes 16-31 -> M=r+8.
// All 16 N-columns of D are identical (B replicated), so acc[r] on lane l is
// s_{r + (l>=16 ? 8 : 0)}.  Redistribute so lane l ends with s_{l % 16}.
__device__ __forceinline__ float ug_extract_score(v8f acc, int lane) {
    int j = lane & 7;
    float t0 = (j & 1) ? acc[1] : acc[0];
    float t1 = (j & 1) ? acc[3] : acc[2];
    float t2 = (j & 1) ? acc[5] : acc[4];
    float t3 = (j & 1) ? acc[7] : acc[6];
    float u0 = (j & 2) ? t1 : t0;
    float u1 = (j & 2) ? t3 : t2;
    float x  = (j & 4) ? u1 : u0;      // x = s_{(lane&7) + (lane>=16 ? 8 : 0)}
    int src = (lane >= 8 && lane < 24) ? (lane ^ 24) : lane;
    return __shfl(x, src, 32);         // lane l now holds s_{l % 16}
}

// One 16-row tile: D = A(16x64) * ue(64), two independent K-chains for ILP.
__device__ __forceinline__ v8f ug_tile_scores(const float* __restrict__ row,
                                              const v2f* __restrict__ ub,
                                              int koff) {
    v8f acc0 = {0.f, 0.f, 0.f, 0.f, 0.f, 0.f, 0.f, 0.f};
    v8f acc1 = {0.f, 0.f, 0.f, 0.f, 0.f, 0.f, 0.f, 0.f};
#pragma unroll
    for (int t = 0; t < 16; t += 2) {
        v2f a0 = *(const v2f*)(row + 4 * t + koff);
        v2f a1 = *(const v2f*)(row + 4 * (t + 1) + koff);
        acc0 = __builtin_amdgcn_wmma_f32_16x16x4_f32(
                   false, a0, false, ub[t],     (short)0, acc0, false, false);
        acc1 = __builtin_amdgcn_wmma_f32_16x16x4_f32(
                   false, a1, false, ub[t + 1], (short)0, acc1, false, false);
    }
    return acc0 + acc1;
}

__global__ void __launch_bounds__(256)
ug_main_loss_kernel(const int* __restrict__ users,
                    const int* __restrict__ pos_items,
                    const int* __restrict__ neg_items,
                    const float* __restrict__ user_table,
                    const float* __restrict__ item_table,
                    const float* __restrict__ beta_uD,
                    const float* __restrict__ beta_iD,
                    const int* __restrict__ nbr_mat,
                    const float* __restrict__ sim_mat,
                    float* __restrict__ partials) {
    const int lane = threadIdx.x & 31;
    const int wave = threadIdx.x >> 5;
    const int b    = blockIdx.x * 8 + wave;        // 512*8 == 4096 exactly
    const int r    = lane & 15;                    // A-matrix row this lane feeds
    const int koff = (lane < 16) ? 0 : 2;          // K sub-chunk per lane half

    const int   u    = users[b];
    const int   pidx = pos_items[b];
    const float bu   = beta_uD[u];

    // Prefetch all per-lane negative indices (one clause, no per-tile stall).
    const int* negb = neg_items + (size_t)b * UG_NUM_NEG;
    int nidx[UG_NEG_TILES];
#pragma unroll
    for (int tt = 0; tt < UG_NEG_TILES; ++tt) {
        int nn = tt * 16 + r;
        if (nn > UG_NUM_NEG - 1) nn = UG_NUM_NEG - 1;   // only tile 18 clamps
        nidx[tt] = negb[nn];
    }

    // B operand: ue[b] K-chunks, replicated across all 16 N columns.
    // Inferred B layout (mirrors C/D): VGPR0 -> K=0 (lanes 0-15) / K=2 (16-31),
    // VGPR1 -> K=1 / K=3.
    const float* ue = user_table + (size_t)u * UG_DIM;
    v2f ub[16];
#pragma unroll
    for (int t = 0; t < 16; ++t)
        ub[t] = *(const v2f*)(ue + 4 * t + koff);

    float csum = 0.0f;

    // ---- negative tiles 0..17 (always 16 valid rows), two tiles in flight ----
    // Fully unrolled: nidx[] becomes named registers (no v_movrels / m0), so the
    // scheduler can hoist tile B's address math + loads into tile A's WMMA shadow.
#pragma unroll
    for (int tp = 0; tp < 9; ++tp) {
        const float* rowA = item_table + (size_t)nidx[2 * tp]     * UG_DIM;
        const float* rowB = item_table + (size_t)nidx[2 * tp + 1] * UG_DIM;

        v8f aA0 = {0.f,0.f,0.f,0.f,0.f,0.f,0.f,0.f};
        v8f aA1 = {0.f,0.f,0.f,0.f,0.f,0.f,0.f,0.f};
        v8f aB0 = {0.f,0.f,0.f,0.f,0.f,0.f,0.f,0.f};
        v8f aB1 = {0.f,0.f,0.f,0.f,0.f,0.f,0.f,0.f};
#pragma unroll
        for (int t = 0; t < 16; t += 2) {
            v2f xA0 = *(const v2f*)(rowA + 4 * t + koff);
            v2f xA1 = *(const v2f*)(rowA + 4 * (t + 1) + koff);
            v2f xB0 = *(const v2f*)(rowB + 4 * t + koff);
            v2f xB1 = *(const v2f*)(rowB + 4 * (t + 1) + koff);
            aA0 = __builtin_amdgcn_wmma_f32_16x16x4_f32(
                      false, xA0, false, ub[t],     (short)0, aA0, false, false);
            aB0 = __builtin_amdgcn_wmma_f32_16x16x4_f32(
                      false, xB0, false, ub[t],     (short)0, aB0, false, false);
            aA1 = __builtin_amdgcn_wmma_f32_16x16x4_f32(
                      false, xA1, false, ub[t + 1], (short)0, aA1, false, false);
            aB1 = __builtin_amdgcn_wmma_f32_16x16x4_f32(
                      false, xB1, false, ub[t + 1], (short)0, aB1, false, false);
        }
        const float sA = ug_extract_score(aA0 + aA1, lane);
        const float sB = ug_extract_score(aB0 + aB1, lane);
        // wce(labels=0): logits + log_term(logits); neg_w is dead code.
        csum += (sA + ug_log_term(sA)) + (sB + ug_log_term(sB));
    }

    // ---- negative tile 18: base 288, 12 valid rows ----
    {
        const float* row = item_table + (size_t)nidx[18] * UG_DIM;
        const float s = ug_extract_score(ug_tile_scores(row, ub, koff), lane);
        csum += (r < UG_NUM_NEG - 288) ? (s + ug_log_term(s)) : 0.0f;
    }

    // ---- pos item + 10 neighbors in one final tile ----
    {
        int idx; float w;
        if (r == 0) {
            idx = pidx;
            w   = UG_W1 + bu * beta_iD[pidx];           // pos_w, labels = 1
        } else if (r <= UG_NUM_NBR) {
            idx = nbr_mat[(size_t)pidx * UG_NUM_NBR + (r - 1)];
            w   = UG_LAMBDA * sim_mat[(size_t)pidx * UG_NUM_NBR + (r - 1)];
        } else {
            idx = pidx;                                 // pad, zero weight
            w   = 0.0f;
        }
        const float* row = item_table + (size_t)idx * UG_DIM;
        const float s = ug_extract_score(ug_tile_scores(row, ub, koff), lane);
        // pos: pos_w * log_term(s) ; nbr: LAMBDA * sim * log_term(s)
        csum += w * ug_log_term(s);
    }

    // Each score is held by 2 lanes -> halve, then wave32 reduce.
    float tsum = csum * 0.5f;
#pragma unroll
    for (int off = 16; off > 0; off >>= 1)
        tsum += __shfl_xor(tsum, off, 32);

    __shared__ float wsum[8];
    if (lane == 0) wsum[wave] = tsum;
    __syncthreads();
    if (threadIdx.x == 0) {
        float blk = 0.0f;
#pragma unroll
        for (int i = 0; i < 8; ++i) blk += wsum[i];
        partials[blockIdx.x] = blk;
    }
}

__global__ void __launch_bounds__(256)
ug_norm_kernel(const float* __restrict__ user_table,
               const float* __restrict__ item_table,
               float* __restrict__ partials) {
    const size_t NU = (size_t)100000 * UG_DIM / 4;   // float4 counts
    const size_t NI = (size_t)50000  * UG_DIM / 4;
    const size_t tid    = (size_t)blockIdx.x * blockDim.x + threadIdx.x;
    const size_t stride = (size_t)gridDim.x * blockDim.x;

    float acc = 0.0f;
    const float4* up = (const float4*)user_table;
    for (size_t i = tid; i < NU; i += stride) {
        float4 v = up[i];
        acc += v.x * v.x + v.y * v.y + v.z * v.z + v.w * v.w;
    }
    const float4* ip = (const float4*)item_table;
    for (size_t i = tid; i < NI; i += stride) {
        float4 v = ip[i];
        acc += v.x * v.x + v.y * v.y + v.z * v.z + v.w * v.w;
    }

    __shared__ float sh[256];
    sh[threadIdx.x] = acc;
    __syncthreads();
    for (int s = 128; s > 0; s >>= 1) {
        if ((int)threadIdx.x < s) sh[threadIdx.x] += sh[threadIdx.x + s];
        __syncthreads();
    }
    if (threadIdx.x == 0) partials[blockIdx.x] = sh[0];
}

__global__ void __launch_bounds__(256)
ug_final_kernel(const float* __restrict__ mainP, int nMain,
                const float* __restrict__ normP, int nNorm,
                float* __restrict__ out) {
    __shared__ double shA[256];
    __shared__ double shN[256];
    double a = 0.0, n = 0.0;
    for (int i = threadIdx.x; i < nMain; i += 256) a += (double)mainP[i];
    for (int i = threadIdx.x; i < nNorm; i += 256) n += (double)normP[i];
    shA[threadIdx.x] = a;
    shN[threadIdx.x] = n;
    __syncthreads();
    for (int s = 128; s > 0; s >>= 1) {
        if ((int)threadIdx.x < s) {
            shA[threadIdx.x] += shA[threadIdx.x + s];
            shN[threadIdx.x] += shN[threadIdx.x + s];
        }
        __syncthreads();
    }
    if (threadIdx.x == 0)
        out[0] = (float)(shA[0] + 0.5 * (double)UG_GAMMA * shN[0]);
}

extern "C" void kernel_launch(void* const* d_in, const int* in_sizes, int n_in,
                              void* d_out, int out_size, void* d_ws, size_t ws_size,
                              hipStream_t stream) {
    (void)in_sizes; (void)n_in; (void)out_size; (void)ws_size;

    const int*   users      = (const int*)  d_in[0];
    const int*   pos_items  = (const int*)  d_in[1];
    const int*   neg_items  = (const int*)  d_in[2];
    const float* user_table = (const float*)d_in[3];
    const float* item_table = (const float*)d_in[4];
    const float* beta_uD    = (const float*)d_in[5];
    const float* beta_iD    = (const float*)d_in[6];
    const int*   nbr_mat    = (const int*)  d_in[7];
    const float* sim_mat    = (const float*)d_in[8];

    float* mainP = (float*)d_ws;                  // [0, 512)
    float* normP = (float*)d_ws + UG_MAIN_BLOCKS; // [512, 512+600)

    ug_main_loss_kernel<<<UG_MAIN_BLOCKS, 256, 0, stream>>>(
        users, pos_items, neg_items, user_table, item_table,
        beta_uD, beta_iD, nbr_mat, sim_mat, mainP);

    ug_norm_kernel<<<UG_NORM_BLOCKS, 256, 0, stream>>>(
        user_table, item_table, normP);

    ug_final_kernel<<<1, 256, 0, stream>>>(
        mainP, UG_MAIN_BLOCKS, normP, UG_NORM_BLOCKS, (float*)d_out);
}